// GyroLoss_61280593379804
// MI455X (gfx1250) — compile-verified
//
#include <hip/hip_runtime.h>
#include <math.h>

// ---------------------------------------------------------------------------
// GyroLoss fused kernel for MI455X (gfx1250, wave32).
//  - one pass over ~25 MB of input (bandwidth-bound: ~1.1us floor @ 23.3TB/s)
//  - hat_xs staged through LDS with CDNA5 async global->LDS (ASYNCcnt path),
//    fully coalesced b128 global traffic
//  - entire 16-matrix SO(3) chain held in registers per thread
//  - level-5 (pair) products via __shfl_xor across adjacent wave32 lanes
//  - 32-lane loss reduction done with one V_WMMA_F32_16X16X4_F32 (ones trick)
// ---------------------------------------------------------------------------

typedef __attribute__((ext_vector_type(2))) float v2f;
typedef __attribute__((ext_vector_type(8))) float v8f;

#define GYRO_DT   0.005f
#define GYRO_EPS  1e-7f
#define GYRO_HUB  0.005f

// W * HUBER^2 / (32*2043*3)  and  W * HUBER^2 / (32*1019*3) / 2
#define GYRO_C1   (2.5e-5f / 196128.0f)
#define GYRO_C2   (2.5e-5f / 97824.0f * 0.5f)

#define LDS_ROW   52          // floats per thread row (208 B, 16B aligned, padded)

__device__ __forceinline__ void so3_exp3(float px, float py, float pz, float* R) {
  float a2    = px * px + py * py + pz * pz;
  float angle = sqrtf(a2);
  if (angle < GYRO_EPS) {
    R[0] = 1.0f; R[1] = -pz;  R[2] = py;
    R[3] = pz;   R[4] = 1.0f; R[5] = -px;
    R[6] = -py;  R[7] = px;   R[8] = 1.0f;
  } else {
    float s, c;
    __sincosf(angle, &s, &c);
    float inv = 1.0f / angle;
    float ax = px * inv, ay = py * inv, az = pz * inv;
    float t = 1.0f - c;
    R[0] = c + t * ax * ax;     R[1] = t * ax * ay - s * az; R[2] = t * ax * az + s * ay;
    R[3] = t * ax * ay + s * az; R[4] = c + t * ay * ay;     R[5] = t * ay * az - s * ax;
    R[6] = t * ax * az - s * ay; R[7] = t * ay * az + s * ax; R[8] = c + t * az * az;
  }
}

__device__ __forceinline__ void mm3(const float* A, const float* B, float* C) {
#pragma unroll
  for (int i = 0; i < 3; ++i)
#pragma unroll
    for (int j = 0; j < 3; ++j)
      C[3 * i + j] = A[3 * i + 0] * B[0 + j] + A[3 * i + 1] * B[3 + j] + A[3 * i + 2] * B[6 + j];
}

// C = A^T * B   (einsum nij,nik->njk)
__device__ __forceinline__ void mtm3(const float* A, const float* B, float* C) {
#pragma unroll
  for (int i = 0; i < 3; ++i)
#pragma unroll
    for (int j = 0; j < 3; ++j)
      C[3 * i + j] = A[i] * B[j] + A[3 + i] * B[3 + j] + A[6 + i] * B[6 + j];
}

__device__ __forceinline__ void so3_log3(const float* M, float* r) {
  float tr = M[0] + M[4] + M[8];
  float ca = fminf(1.0f, fmaxf(-1.0f, 0.5f * tr - 0.5f));
  float angle = acosf(ca);
  if (angle < GYRO_EPS) {
    r[0] = M[7]; r[1] = M[2]; r[2] = M[3];          // vee(M - I)
  } else {
    float coef = 0.5f * angle / sinf(angle);
    r[0] = coef * (M[7] - M[5]);
    r[1] = coef * (M[2] - M[6]);
    r[2] = coef * (M[3] - M[1]);
  }
}

__device__ __forceinline__ float huber3(const float* r) {
  const float inv_h = 1.0f / GYRO_HUB;
  float s = 0.0f;
#pragma unroll
  for (int c = 0; c < 3; ++c) {
    float x = fabsf(r[c]) * inv_h;
    s += (x < 1.0f) ? 0.5f * x * x : x - 0.5f;
  }
  return s;
}

__global__ void GyroLoss_zero(float* out) { out[0] = 0.0f; }

__global__ void __launch_bounds__(256)
GyroLoss_kernel(const float* __restrict__ xs, const float* __restrict__ hat_xs,
                float* __restrict__ out) {
  const int g     = blockIdx.x * 256 + threadIdx.x;  // group-of-16 index, [0, 65536)
  const int idx16 = g & 2047;                        // position within batch (2048 groups)

  // -------------------------------------------------------------------------
  // Stage this block's 48KB hat_xs tile into LDS with async global->LDS DMA.
  // Chunk c (16B) of the tile -> owner thread c/12, slot c%12, padded rows so
  // per-thread LDS reads avoid heavy bank conflicts. Global side is fully
  // coalesced: consecutive lanes fetch consecutive 16B chunks.
  // -------------------------------------------------------------------------
  __shared__ __align__(16) float tile[256 * LDS_ROW];
  {
    const uint32_t lds_base = (uint32_t)(uintptr_t)&tile[0];
    const uint32_t goff0    = (uint32_t)blockIdx.x * 49152u;   // block tile, bytes
#pragma unroll
    for (int k = 0; k < 12; ++k) {
      int c      = (int)threadIdx.x + k * 256;                 // chunk index in tile
      int owner  = c / 12;
      int slot   = c - owner * 12;
      uint32_t lds_b = lds_base + (uint32_t)owner * (LDS_ROW * 4u) + (uint32_t)slot * 16u;
      uint32_t voff  = goff0 + (uint32_t)c * 16u;
      asm volatile("global_load_async_to_lds_b128 %0, %1, %2"
                   :
                   : "v"(lds_b), "v"(voff), "s"(hat_xs)
                   : "memory");
    }
    asm volatile("s_wait_asynccnt 0x0" ::: "memory");
    __syncthreads();
  }

  // ---- read this thread's 16 gyro samples (48 f32) from LDS ----
  float w[48];
  {
    const float* myrow = &tile[threadIdx.x * LDS_ROW];
#pragma unroll
    for (int i = 0; i < 12; ++i) {
      float4 v = *(const float4*)(myrow + 4 * i);
      w[4 * i + 0] = v.x; w[4 * i + 1] = v.y; w[4 * i + 2] = v.z; w[4 * i + 3] = v.w;
    }
  }

  // ---- P = prod_{i=0..15} so3_exp(DT * w_i), ordered left-to-right ----
  float P[9], O[9], T[9];
  so3_exp3(GYRO_DT * w[0], GYRO_DT * w[1], GYRO_DT * w[2], P);
#pragma unroll
  for (int i = 1; i < 16; ++i) {
    so3_exp3(GYRO_DT * w[3 * i], GYRO_DT * w[3 * i + 1], GYRO_DT * w[3 * i + 2], O);
    mm3(P, O, T);
#pragma unroll
    for (int k = 0; k < 9; ++k) P[k] = T[k];
  }

  // ---- X = so3_exp(xs[b, idx16*16]) (the ::16 subsample) ----
  float X[9];
  {
    const float* xp = xs + (size_t)g * 48;  // b*32768*3 + idx16*16*3 == g*48
    so3_exp3(xp[0], xp[1], xp[2], X);
  }

  // ---- level-4 (MIN_N) residual ----
  float M1[9], r1[3];
  mtm3(P, X, M1);
  so3_log3(M1, r1);
  float l1 = (idx16 >= 5) ? huber3(r1) : 0.0f;

  // ---- level-5: pair adjacent lanes (g even takes g,g+1) via wave32 shuffles ----
  float Pp[9], Xp[9];
#pragma unroll
  for (int k = 0; k < 9; ++k) {
    Pp[k] = __shfl_xor(P[k], 1, 32);
    Xp[k] = __shfl_xor(X[k], 1, 32);
  }
  float l2;
  {
    float P32[9], X32[9], M2[9], r2[3];
    mm3(P, Pp, P32);
    mm3(X, Xp, X32);
    mtm3(P32, X32, M2);
    so3_log3(M2, r2);
    const int  idx32  = idx16 >> 1;
    const bool active = ((g & 1) == 0) && (idx32 >= 5);
    l2 = active ? huber3(r2) : 0.0f;
  }

  // ---- per-lane scaled contribution, then 32-lane sum via one WMMA ----
  // A(16x4): lane L -> A[L%16][L<16 ? 0 : 2] = v, other K slots 0. B(4x16)=ones.
  // D[m][n] = v[m] + v[m+16]; sum of D's 8 VGPRs + xor-16 fold = full wave sum.
  float v = l1 * GYRO_C1 + l2 * GYRO_C2;
  v2f a;  a.x = v;    a.y = 0.0f;
  v2f bo; bo.x = 1.0f; bo.y = 1.0f;
  v8f c = {};
  v8f d = __builtin_amdgcn_wmma_f32_16x16x4_f32(false, a, false, bo, (short)0, c,
                                                false, false);
  float t = d[0] + d[1] + d[2] + d[3] + d[4] + d[5] + d[6] + d[7];
  t += __shfl_xor(t, 16, 32);   // now every lane holds the wave-wide sum

  // ---- block reduction (8 waves) + one atomic per block ----
  __shared__ float wsum[8];
  const int wid  = threadIdx.x >> 5;
  const int lane = threadIdx.x & 31;
  if (lane == 0) wsum[wid] = t;
  __syncthreads();
  if (threadIdx.x == 0) {
    float s = 0.0f;
#pragma unroll
    for (int i = 0; i < 8; ++i) s += wsum[i];
    atomicAdd(out, s);
  }
}

extern "C" void kernel_launch(void* const* d_in, const int* in_sizes, int n_in,
                              void* d_out, int out_size, void* d_ws, size_t ws_size,
                              hipStream_t stream) {
  const float* xs     = (const float*)d_in[0];
  const float* hat_xs = (const float*)d_in[1];
  float*       out    = (float*)d_out;

  // groups of 16 samples; reference shape: 32 x 32768 x 3 -> 65536 groups
  const int total  = in_sizes[1] / 3;   // 1048576 samples
  const int g16    = total / 16;        // 65536 threads
  const int blocks = g16 / 256;         // 256 blocks of 256 threads (8 wave32s)

  GyroLoss_zero<<<1, 1, 0, stream>>>(out);
  GyroLoss_kernel<<<blocks, 256, 0, stream>>>(xs, hat_xs, out);
}